// ProjectionLayer_1554778161473
// MI455X (gfx1250) — compile-verified
//
#include <hip/hip_runtime.h>

typedef __attribute__((ext_vector_type(2))) float v2f;
typedef __attribute__((ext_vector_type(8))) float v8f;

#define POSE_SCALING 0.01f
#define BATCH 16
#define HH    480
#define WW    640
#define HWPIX (HH * WW)   // 307200, multiple of 256 -> each block within one batch

// D = A(16x4) * B(4x16) + C using V_WMMA_F32_16X16X4_F32.
// A rows 0..2 = M4 = [K*R*Kinv | K*t]  (3x4 affine, rest zero)
// B columns   = q = (px*d, py*d, d, 1) for 16 pixels
// D rows 0..2 = (X, Y, Z) projected homogeneous coords per pixel.

__global__ __launch_bounds__(256) void ProjectionLayer_kernel(
    const float* __restrict__ src,    // [B,H,W,3]
    const float* __restrict__ depth,  // [B,H,W,1]
    const float* __restrict__ pose,   // [B,6]
    float* __restrict__ out)          // [B,H,W,3]
{
    __shared__ float sM[12];          // M4 row-major 3x4

    const int tid = threadIdx.x;
    const int g   = blockIdx.x * 256 + tid;      // global pixel id
    const int b   = g / HWPIX;
    const int rem = g % HWPIX;
    const int y   = rem / WW;
    const int x   = rem % WW;

    // ---- per-block pose -> M4 = [K*R*Kinv | K*t] ----
    if (tid == 0) {
        const float* p = pose + b * 6;
        const float t0 = p[0] * POSE_SCALING;
        const float t1 = p[1] * POSE_SCALING;
        const float t2 = p[2] * POSE_SCALING;
        const float PI = 3.14159265358979323846f;
        const float rx = fminf(fmaxf(p[3] * POSE_SCALING, -PI), PI);
        const float ry = fminf(fmaxf(p[4] * POSE_SCALING, -PI), PI);
        const float rz = fminf(fmaxf(p[5] * POSE_SCALING, -PI), PI);
        const float cx = cosf(rx), sx = sinf(rx);
        const float cy = cosf(ry), sy = sinf(ry);
        const float cz = cosf(rz), sz = sinf(rz);
        // R = Rx @ Ry @ Rz
        const float R00 = cy * cz,                 R01 = -cy * sz,                R02 = sy;
        const float R10 = cx * sz + sx * sy * cz,  R11 = cx * cz - sx * sy * sz,  R12 = -sx * cy;
        const float R20 = sx * sz - cx * sy * cz,  R21 = sx * cz + cx * sy * sz,  R22 = cx * cy;
        // G = K @ R   (K adds 0.5*row2 to rows 0,1)
        const float G00 = R00 + 0.5f * R20, G01 = R01 + 0.5f * R21, G02 = R02 + 0.5f * R22;
        const float G10 = R10 + 0.5f * R20, G11 = R11 + 0.5f * R21, G12 = R12 + 0.5f * R22;
        const float G20 = R20,              G21 = R21,              G22 = R22;
        // A3 = G @ Kinv  (col2 -= 0.5*(col0+col1))
        sM[0]  = G00; sM[1]  = G01; sM[2]  = G02 - 0.5f * (G00 + G01); sM[3]  = t0 + 0.5f * t2;
        sM[4]  = G10; sM[5]  = G11; sM[6]  = G12 - 0.5f * (G10 + G11); sM[7]  = t1 + 0.5f * t2;
        sM[8]  = G20; sM[9]  = G21; sM[10] = G22 - 0.5f * (G20 + G21); sM[11] = t2;
    }
    __syncthreads();

    const int  lane = tid & 31;
    const int  m    = lane & 15;
    const bool hi   = lane >= 16;

    // ---- A matrix (16x4 f32): lanes 0-15 hold K=0,1 ; lanes 16-31 hold K=2,3 ----
    v2f a;
    a.x = (m < 3) ? sM[m * 4 + (hi ? 2 : 0)] : 0.0f;
    a.y = (m < 3) ? sM[m * 4 + (hi ? 3 : 1)] : 0.0f;

    // ---- per-pixel homogeneous point q ----
    const float d   = depth[g];
    const float pxn = ((float)x + 0.5f) * (1.0f / (float)WW);
    const float pyn = ((float)y + 0.5f) * (1.0f / (float)HH);
    const float q0  = pxn * d;
    const float q1  = pyn * d;

    // ---- B group 0: pixels on lanes 0-15 ----
    // low lanes: rows K=0,1 -> (q0,q1); high lanes: rows K=2,3 -> (d_of_lane-16, 1)
    const float d_lo = __shfl(d, m, 32);
    v2f b0;
    b0.x = hi ? d_lo : q0;
    b0.y = hi ? 1.0f : q1;

    // ---- B group 1: pixels on lanes 16-31 ----
    const float q0_hi = __shfl(q0, m + 16, 32);
    const float q1_hi = __shfl(q1, m + 16, 32);
    v2f b1;
    b1.x = hi ? d    : q0_hi;
    b1.y = hi ? 1.0f : q1_hi;

    // ---- two WMMAs: D rows 0..2 = X,Y,Z per pixel column ----
    v8f c0 = {};
    c0 = __builtin_amdgcn_wmma_f32_16x16x4_f32(false, a, false, b0, (short)0, c0, false, false);
    v8f c1 = {};
    c1 = __builtin_amdgcn_wmma_f32_16x16x4_f32(false, a, false, b1, (short)0, c1, false, false);

    // results for pixels 16-31 sit in lanes 0-15 of c1 -> shuffle up to high lanes
    const float X1 = __shfl(c1[0], m, 32);
    const float Y1 = __shfl(c1[1], m, 32);
    const float Z1 = __shfl(c1[2], m, 32);
    const float X  = hi ? X1 : c0[0];
    const float Y  = hi ? Y1 : c0[1];
    const float Z  = hi ? Z1 : c0[2];

    // ---- perspective divide: v_rcp_f32 + one Newton-Raphson step ----
    // Z ~ depth in (0.1, 1.1): well-conditioned; NR step restores ~full f32 precision.
    const float zd   = Z + 1e-10f;
    float zinv = __builtin_amdgcn_rcpf(zd);
    zinv = zinv * fmaf(-zd, zinv, 2.0f);
    const float u = (X * zinv) * (float)WW - 0.5f;
    const float v = (Y * zinv) * (float)HH - 0.5f;

    // ---- bilinear sample with zero padding ----
    const float x0f = floorf(u), y0f = floorf(v);
    const float wx = u - x0f,    wy = v - y0f;
    const int x0 = (int)x0f,     y0 = (int)y0f;
    const int x1 = x0 + 1,       y1 = y0 + 1;
    const bool valid = (x0 >= 0) && (x1 <= WW - 1) && (y0 >= 0) && (y1 <= HH - 1);
    const int x0c = min(max(x0, 0), WW - 1);
    const int x1c = min(max(x1, 0), WW - 1);
    const int y0c = min(max(y0, 0), HH - 1);
    const int y1c = min(max(y1, 0), HH - 1);

    const float w00 = (1.0f - wx) * (1.0f - wy);
    const float w01 = wx * (1.0f - wy);
    const float w10 = (1.0f - wx) * wy;
    const float w11 = wx * wy;

    const float* img = src + (size_t)b * HWPIX * 3;
    const size_t i00 = ((size_t)y0c * WW + x0c) * 3;
    const size_t i01 = ((size_t)y0c * WW + x1c) * 3;
    const size_t i10 = ((size_t)y1c * WW + x0c) * 3;
    const size_t i11 = ((size_t)y1c * WW + x1c) * 3;

    const float scale = valid ? 1.0f : 0.0f;
    float* o = out + (size_t)g * 3;
#pragma unroll
    for (int c = 0; c < 3; ++c) {
        const float r = img[i00 + c] * w00 + img[i01 + c] * w01 +
                        img[i10 + c] * w10 + img[i11 + c] * w11;
        o[c] = r * scale;
    }
}

extern "C" void kernel_launch(void* const* d_in, const int* in_sizes, int n_in,
                              void* d_out, int out_size, void* d_ws, size_t ws_size,
                              hipStream_t stream) {
    const float* src   = (const float*)d_in[0];  // source_img [16,480,640,3]
    const float* depth = (const float*)d_in[1];  // depth_map  [16,480,640,1]
    const float* pose  = (const float*)d_in[2];  // pose       [16,6]
    float* out = (float*)d_out;                  // [16,480,640,3]

    const int total  = BATCH * HWPIX;            // 4,915,200 pixels
    const int blocks = total / 256;              // 19,200 (exact; EXEC all-ones for WMMA)
    ProjectionLayer_kernel<<<blocks, 256, 0, stream>>>(src, depth, pose, out);
}